// Linear_89696097009734
// MI455X (gfx1250) — compile-verified
//
#include <hip/hip_runtime.h>

typedef __attribute__((ext_vector_type(2))) float v2f;
typedef __attribute__((ext_vector_type(8))) float v8f;

#define NPED 4096
#define NSLAB 16                      // j-range split for occupancy
#define JSPAN (NPED / NSLAB)          // 256 j's per slab = 16 j-tiles

// exp((0.6 - d)/0.71) == exp2((0.6 - d) * log2(e)/0.71)
#define EXP2_SCALE 2.0319648463224837f
#define REP_ALPHA 10.66f

__global__ void zero_ws_kernel(float* __restrict__ ws, int n) {
  int t = blockIdx.x * blockDim.x + threadIdx.x;
  if (t < n) ws[t] = 0.0f;
}

// One wave (32 threads) per block. Block = (i-tile, j-slab).
// Accumulates per-i (Tx, Ty, S) = (Σ w*xj, Σ w*yj, Σ w) via f32 WMMA,
// drains via global f32 atomics into acc3[3*N].
__global__ __launch_bounds__(32) void pair_force_wmma(
    const float* __restrict__ state, float* __restrict__ acc3) {
  const int lane = threadIdx.x;       // 0..31
  const int n    = lane & 15;         // N column / A row within tile
  const int half = lane >> 4;         // 0 or 1 (K group)
  const int it   = blockIdx.x;        // i-tile 0..255
  const int j0   = blockIdx.y * JSPAN;

  const int   i  = it * 16 + n;       // this lane's A-matrix row (M = n)
  const float xi = state[i * 4 + 0];
  const float yi = state[i * 4 + 1];

  // per-lane B-column selectors: column 0 -> xj, 1 -> yj, 2 -> 1.0, else 0
  const float s0 = (n == 0) ? 1.0f : 0.0f;
  const float s1 = (n == 1) ? 1.0f : 0.0f;
  const float s2 = (n == 2) ? 1.0f : 0.0f;

  v8f accA = {};                      // ping-pong accumulators to break the
  v8f accB = {};                      // WMMA D->C dependency chain

  for (int jt = 0; jt < JSPAN; jt += 16) {
#pragma unroll
    for (int c = 0; c < 4; ++c) {
      // K chunk c covers j = base .. base+3; this lane owns K = 2*half, 2*half+1
      const int j = j0 + jt + 4 * c + 2 * half;
      const float2 pj0 = *reinterpret_cast<const float2*>(state + j * 4);
      const float2 pj1 = *reinterpret_cast<const float2*>(state + j * 4 + 4);

      float dx0 = xi - pj0.x, dy0 = yi - pj0.y;
      float dx1 = xi - pj1.x, dy1 = yi - pj1.y;
      float q0 = dx0 * dx0 + dy0 * dy0 + 1e-8f;
      float q1 = dx1 * dx1 + dy1 * dy1 + 1e-8f;
      float r0 = __builtin_amdgcn_rsqf(q0);          // v_rsq_f32
      float r1 = __builtin_amdgcn_rsqf(q1);
      float d0 = q0 * r0;                            // dist = q * rsqrt(q)
      float d1 = q1 * r1;
      float w0 = REP_ALPHA * __builtin_amdgcn_exp2f((0.6f - d0) * EXP2_SCALE) * r0;
      float w1 = REP_ALPHA * __builtin_amdgcn_exp2f((0.6f - d1) * EXP2_SCALE) * r1;
      w0 = (i == j)     ? 0.0f : w0;                 // diagonal mask
      w1 = (i == j + 1) ? 0.0f : w1;

      v2f a;  a.x = w0;  a.y = w1;    // A[M=n][K=2h], A[M=n][K=2h+1]
      v2f b;                          // B[K=2h][N=n], B[K=2h+1][N=n]
      b.x = __builtin_fmaf(s0, pj0.x, __builtin_fmaf(s1, pj0.y, s2));
      b.y = __builtin_fmaf(s0, pj1.x, __builtin_fmaf(s1, pj1.y, s2));

      if (c & 1)
        accB = __builtin_amdgcn_wmma_f32_16x16x4_f32(
            false, a, false, b, (short)0, accB, false, false);
      else
        accA = __builtin_amdgcn_wmma_f32_16x16x4_f32(
            false, a, false, b, (short)0, accA, false, false);
    }
  }

  const v8f acc = accA + accB;        // identical C/D layouts -> elementwise add

  // D layout: VGPR r holds M=r (lanes 0-15) and M=r+8 (lanes 16-31), N = lane&15.
  if (n < 3) {
    const int mbase = it * 16 + half * 8;
#pragma unroll
    for (int r = 0; r < 8; ++r)
      atomicAdd(&acc3[3 * (mbase + r) + n], acc[r]);
  }
}

__global__ void finalize_kernel(const float* __restrict__ state,
                                const float* __restrict__ cost_in,
                                const float* __restrict__ goals,
                                const float* __restrict__ rip,
                                const float* __restrict__ obs,
                                const float* __restrict__ acc3,
                                float* __restrict__ out_state,
                                float* __restrict__ out_cost) {
  const int i = blockIdx.x * blockDim.x + threadIdx.x;
  if (i >= NPED) return;
  const float EPS = 1e-8f, VMAX = 1.34f;

  auto propagate = [&](int idx, float& px, float& py, float& vx, float& vy) {
    float x  = state[idx * 4 + 0], y  = state[idx * 4 + 1];
    float ux = state[idx * 4 + 2], uy = state[idx * 4 + 3];
    float Tx = acc3[idx * 3 + 0], Ty = acc3[idx * 3 + 1], S = acc3[idx * 3 + 2];
    float rfx = x * S - Tx, rfy = y * S - Ty;          // repulsive force
    float tgx = goals[idx * 2 + 0] - x, tgy = goals[idx * 2 + 1] - y;
    float rgd = __builtin_amdgcn_rsqf(tgx * tgx + tgy * tgy + EPS);
    float afx = 120.0f * (VMAX * tgx * rgd - ux);      // K*M = 2*60
    float afy = 120.0f * (VMAX * tgy * rgd - uy);
    float Fx = rfx + afx, Fy = rfy + afy;
    float nvx = ux + Fx * (0.4f / 60.0f);              // DT/PED_MASS
    float nvy = uy + Fy * (0.4f / 60.0f);
    float sp = sqrtf(nvx * nvx + nvy * nvy + EPS);
    float sc = fminf(1.0f, VMAX / sp);
    nvx *= sc; nvy *= sc;
    px = x + nvx * 0.4f; py = y + nvy * 0.4f;
    vx = nvx; vy = nvy;
  };

  float px, py, vx, vy;
  propagate(i, px, py, vx, vy);
  out_state[i * 4 + 0] = px; out_state[i * 4 + 1] = py;
  out_state[i * 4 + 2] = vx; out_state[i * 4 + 3] = vy;

  // robot = pedestrian 0's NEW pose (recomputed redundantly per thread)
  float rpx, rpy, rvx, rvy;
  propagate(0, rpx, rpy, rvx, rvy);

  float gvx = goals[0] - rip[0], gvy = goals[1] - rip[1];
  float pg = ((rpx - rip[0]) * gvx + (rpy - rip[1]) * gvy) /
             (sqrtf(gvx * gvx + gvy * gvy) + EPS);
  float drx = px - rpx, dry = py - rpy;
  float blame = __builtin_amdgcn_exp2f(-1.4426950408889634f *
                                       sqrtf(drx * drx + dry * dry + EPS));
  float dvx = px - obs[i * 4 + 0], dvy = py - obs[i * 4 + 1];
  float dev = dvx * dvx + dvy * dvy;
  out_cost[i] = cost_in[i] + (-pg + blame + dev);
}

extern "C" void kernel_launch(void* const* d_in, const int* in_sizes, int n_in,
                              void* d_out, int out_size, void* d_ws, size_t ws_size,
                              hipStream_t stream) {
  const float* state = (const float*)d_in[0];   // (N,4)
  const float* cost  = (const float*)d_in[1];   // (N,1)
  const float* goals = (const float*)d_in[2];   // (N,2)
  const float* rip   = (const float*)d_in[3];   // (2,)
  const float* obs   = (const float*)d_in[4];   // (N,4)
  float* out_state = (float*)d_out;             // (N,4) flat
  float* out_cost  = (float*)d_out + 4 * NPED;  // (N,1) flat
  float* acc3 = (float*)d_ws;                   // 3*N f32 accumulators

  zero_ws_kernel<<<(3 * NPED + 255) / 256, 256, 0, stream>>>(acc3, 3 * NPED);

  dim3 grid(NPED / 16, NSLAB);                  // 256 i-tiles x 16 j-slabs
  pair_force_wmma<<<grid, 32, 0, stream>>>(state, acc3);

  finalize_kernel<<<(NPED + 255) / 256, 256, 0, stream>>>(
      state, cost, goals, rip, obs, acc3, out_state, out_cost);
}